// CustomTokenEmbeddingModule_56676388438136
// MI455X (gfx1250) — compile-verified
//
#include <hip/hip_runtime.h>

// Embedding gather, fused: out[tok, :] = table_concat[flat_map[ids[tok]], :]
// B=32, S=8192, D=512, VOCAB=1725 (+1 unknown row).
// Memory-bound: 512 MiB NT stores, table (~3.4 MB) stays L2-resident.

#define EMB_D      512
#define EMB_VOCAB  1725

// Native clang vector type so __builtin_nontemporal_store accepts it and
// loads/stores lower to b128 vmem ops.
typedef float v4f __attribute__((ext_vector_type(4)));

// cumulative segment starts (concat order from the reference):
// special[0,3) event[3,12) time[12,524) note[524,652) vel[652,684)
// prog[684,813) local[813,829) ccnum[829,957) ccval[957,1085)
// progval[1085,1213) dur[1213,1725)

__global__ __launch_bounds__(256) void
embed_gather_kernel(const int* __restrict__ ids,
                    const int* __restrict__ flat_map,
                    const float* __restrict__ special_tab,
                    const float* __restrict__ event_tab,
                    const float* __restrict__ time_tab,
                    const float* __restrict__ note_tab,
                    const float* __restrict__ vel_tab,
                    const float* __restrict__ prog_tab,
                    const float* __restrict__ local_tab,
                    const float* __restrict__ ccnum_tab,
                    const float* __restrict__ ccval_tab,
                    const float* __restrict__ progval_tab,
                    const float* __restrict__ dur_tab,
                    const float* __restrict__ unknown_embed,
                    float* __restrict__ out,
                    int ntok)
{
    const int lane = threadIdx.x & 31;            // wave32 lane
    const int wave = threadIdx.x >> 5;            // 8 waves / block
    const int tok  = blockIdx.x * 8 + wave;       // one token row per wave
    if (tok >= ntok) return;

    // Wave-uniform index math (all lanes of this wave share `tok`).
    const int id = ids[tok];
    int fidx = ((unsigned)id < (unsigned)EMB_VOCAB) ? flat_map[id] : EMB_VOCAB;

    const float* src;
    if ((unsigned)fidx >= (unsigned)EMB_VOCAB) {
        src = unknown_embed;                       // out-of-range -> unknown row
    } else if (fidx < 524) {
        if      (fidx < 3)   src = special_tab + (size_t)(fidx        ) * EMB_D;
        else if (fidx < 12)  src = event_tab   + (size_t)(fidx -    3 ) * EMB_D;
        else                 src = time_tab    + (size_t)(fidx -   12 ) * EMB_D;
    } else if (fidx < 829) {
        if      (fidx < 652) src = note_tab    + (size_t)(fidx -  524 ) * EMB_D;
        else if (fidx < 684) src = vel_tab     + (size_t)(fidx -  652 ) * EMB_D;
        else if (fidx < 813) src = prog_tab    + (size_t)(fidx -  684 ) * EMB_D;
        else                 src = local_tab   + (size_t)(fidx -  813 ) * EMB_D;
    } else {
        if      (fidx < 957) src = ccnum_tab   + (size_t)(fidx -  829 ) * EMB_D;
        else if (fidx < 1085)src = ccval_tab   + (size_t)(fidx -  957 ) * EMB_D;
        else if (fidx < 1213)src = progval_tab + (size_t)(fidx - 1085 ) * EMB_D;
        else                 src = dur_tab     + (size_t)(fidx - 1213 ) * EMB_D;
    }

    // 512 floats = 128 v4f chunks; lane l moves chunks l, l+32, l+64, l+96.
    // Consecutive lanes -> contiguous 512B per vmem instruction (fully coalesced).
    const v4f* __restrict__ s4 = (const v4f*)src;
    v4f* __restrict__ d4 = (v4f*)(out + (size_t)tok * EMB_D);

#pragma unroll
    for (int i = 0; i < 4; ++i) {
        const int c = lane + 32 * i;
        v4f v = s4[c];                             // RT load: table stays in L2
        __builtin_nontemporal_store(v, &d4[c]);    // NT store: streaming output
    }
}

extern "C" void kernel_launch(void* const* d_in, const int* in_sizes, int n_in,
                              void* d_out, int out_size, void* d_ws, size_t ws_size,
                              hipStream_t stream) {
    const int*   ids        = (const int*)  d_in[0];   // [B*S] int32
    const int*   flat_map   = (const int*)  d_in[1];   // [VOCAB] int32
    const float* special    = (const float*)d_in[2];   // [3,512]
    const float* event_tab  = (const float*)d_in[3];   // [9,512]
    const float* time_tab   = (const float*)d_in[4];   // [512,512]
    const float* note_tab   = (const float*)d_in[5];   // [128,512]
    const float* vel_tab    = (const float*)d_in[6];   // [32,512]
    const float* prog_tab   = (const float*)d_in[7];   // [129,512]
    const float* local_tab  = (const float*)d_in[8];   // [16,512]
    const float* ccnum_tab  = (const float*)d_in[9];   // [128,512]
    const float* ccval_tab  = (const float*)d_in[10];  // [128,512]
    const float* progval    = (const float*)d_in[11];  // [128,512]
    const float* dur_tab    = (const float*)d_in[12];  // [512,512]
    const float* unknown    = (const float*)d_in[13];  // [512]
    float* out = (float*)d_out;

    const int ntok = in_sizes[0];                 // B*S = 262144
    const int blocks = (ntok + 7) / 8;            // 8 tokens (waves) per block

    embed_gather_kernel<<<blocks, 256, 0, stream>>>(
        ids, flat_map, special, event_tab, time_tab, note_tab, vel_tab,
        prog_tab, local_tab, ccnum_tab, ccval_tab, progval, dur_tab,
        unknown, out, ntok);
}